// FMICLLoss_46273977647737
// MI455X (gfx1250) — compile-verified
//
#include <hip/hip_runtime.h>
#include <hip/hip_bf16.h>
#include <stdint.h>

typedef __attribute__((ext_vector_type(2))) float v2f;
typedef __attribute__((ext_vector_type(8))) float v8f;

#define NROWS 8192
#define DDIM  128
#define BLK   128
#define KCH   32
#define PITCH 36   // 32 + 4 floats pad: 144B row stride (16B aligned), conflict-free frag reads

// Dynamic LDS layout (bytes):
//   As: 2 x 128 x 36 floats   (double-buffered A rows)
//   Bs: 2 x 128 x 36 floats   (double-buffered B cols)
//   sqI[128], sqJ[128], wsum[8]
#define AS_BYTES   (2 * BLK * PITCH * 4)
#define SMEM_BYTES (2 * AS_BYTES + (BLK + BLK + 8) * 4)

// ---------------------------------------------------------------------------
// Kernel 1: L2-normalize rows of z1/z2, positive term partials, store z1n + sq
// One wave32 per row (32 lanes x 4 floats = 128 = D). 8 rows per 256-thr block.
// ---------------------------------------------------------------------------
__global__ __launch_bounds__(256) void fmicl_norm_pos(
    const float* __restrict__ z1, const float* __restrict__ z2,
    float* __restrict__ z1n, float* __restrict__ sq,
    float* __restrict__ pos_partial) {
  const int tid  = threadIdx.x;
  const int lane = tid & 31;
  const int wave = tid >> 5;
  const int row  = blockIdx.x * 8 + wave;
  __shared__ float wsum[8];

  const float4 a = *(const float4*)(z1 + (size_t)row * DDIM + lane * 4);
  const float4 b = *(const float4*)(z2 + (size_t)row * DDIM + lane * 4);

  float s1 = a.x * a.x + a.y * a.y + a.z * a.z + a.w * a.w;
  float s2 = b.x * b.x + b.y * b.y + b.z * b.z + b.w * b.w;
#pragma unroll
  for (int m = 16; m >= 1; m >>= 1) {
    s1 += __shfl_xor(s1, m, 32);
    s2 += __shfl_xor(s2, m, 32);
  }
  const float inv1 = 1.0f / fmaxf(sqrtf(s1), 1e-12f);
  const float inv2 = 1.0f / fmaxf(sqrtf(s2), 1e-12f);

  float4 an = {a.x * inv1, a.y * inv1, a.z * inv1, a.w * inv1};
  float4 bn = {b.x * inv2, b.y * inv2, b.z * inv2, b.w * inv2};
  *(float4*)(z1n + (size_t)row * DDIM + lane * 4) = an;

  const float dx = an.x - bn.x, dy = an.y - bn.y, dz = an.z - bn.z, dw = an.w - bn.w;
  float dpos = dx * dx + dy * dy + dz * dz + dw * dw;
#pragma unroll
  for (int m = 16; m >= 1; m >>= 1) dpos += __shfl_xor(dpos, m, 32);

  if (lane == 0) {
    sq[row] = s1 * inv1 * inv1;                      // ||z1n||^2 as actually stored
    const float g = expf(-0.5f * dpos);              // SIGMA = 1
    wsum[wave] = logf(g + 1e-8f) + 1.0f;             // s_pos
  }
  __syncthreads();
  if (tid == 0) {
    float s = 0.f;
    for (int w = 0; w < 8; ++w) s += wsum[w];
    pos_partial[blockIdx.x] = s;
  }
}

// ---------------------------------------------------------------------------
// Kernel 2: Gram tile via V_WMMA_F32_16X16X4_F32 with double-buffered
// async global->LDS staging (GLOBAL_LOAD_ASYNC_TO_LDS_B128 / ASYNCcnt),
// fused exp(-d2/2) off-diagonal reduction. Upper block-triangle only.
// ---------------------------------------------------------------------------
__global__ __launch_bounds__(256) void fmicl_gram_neg(
    const float* __restrict__ z1n, const float* __restrict__ sq,
    float* __restrict__ neg_partial) {
  const int bi = blockIdx.y;
  const int bj = blockIdx.x;
  const int pidx = bi * 64 + bj;
  const int tid = threadIdx.x;
  if (bj < bi) {                       // symmetric half: counted by (bj,bi) with weight 2
    if (tid == 0) neg_partial[pidx] = 0.f;
    return;
  }

  extern __shared__ __align__(16) char smem_raw[];
  float (*As)[BLK][PITCH] = (float (*)[BLK][PITCH])(smem_raw);
  float (*Bs)[BLK][PITCH] = (float (*)[BLK][PITCH])(smem_raw + AS_BYTES);
  float* sqI  = (float*)(smem_raw + 2 * AS_BYTES);
  float* sqJ  = sqI + BLK;
  float* wsum = sqJ + BLK;

  const int lane    = tid & 31;
  const int wave    = tid >> 5;
  const int m       = lane & 15;  // M (and N) index within 16x16 tile
  const int khalf   = lane >> 4;  // which K-half this lane holds
  const int rowBase = bi * BLK;
  const int colBase = bj * BLK;

  if (tid < BLK) sqI[tid] = sq[rowBase + tid];
  else           sqJ[tid - BLK] = sq[colBase + (tid - BLK)];

  // Per-thread staging coords: 1024 float4 tiles per 128x32 chunk, 4 per thread
  int st_r[4], st_c4[4];
#pragma unroll
  for (int it = 0; it < 4; ++it) {
    const int idx = tid + it * 256;
    st_r[it]  = idx >> 3;
    st_c4[it] = (idx & 7) << 2;
  }

  // Async DMA stage of one 128x32 K-chunk into buffer `buf` (no wait here)
  auto stage = [&](int buf, int kk) {
#pragma unroll
    for (int it = 0; it < 4; ++it) {
      const int r  = st_r[it];
      const int c4 = st_c4[it];
      const float*   ga = z1n + (size_t)(rowBase + r) * DDIM + kk + c4;
      const float*   gb = z1n + (size_t)(colBase + r) * DDIM + kk + c4;
      const uint32_t la = (uint32_t)(uintptr_t)&As[buf][r][c4];
      const uint32_t lb = (uint32_t)(uintptr_t)&Bs[buf][r][c4];
      asm volatile("global_load_async_to_lds_b128 %0, %1, off"
                   :: "v"(la), "v"(ga) : "memory");
      asm volatile("global_load_async_to_lds_b128 %0, %1, off"
                   :: "v"(lb), "v"(gb) : "memory");
    }
  };

  v8f acc[8];
#pragma unroll
  for (int t = 0; t < 8; ++t) acc[t] = (v8f){0.f, 0.f, 0.f, 0.f, 0.f, 0.f, 0.f, 0.f};

  // Prologue: stage chunk 0
  stage(0, 0);
  asm volatile("s_wait_asynccnt 0" ::: "memory");
  __syncthreads();

  for (int c = 0; c < DDIM / KCH; ++c) {
    const int cur = c & 1;
    if (c + 1 < DDIM / KCH) stage(cur ^ 1, (c + 1) * KCH);  // overlap next fetch

#pragma unroll
    for (int ks = 0; ks < KCH / 4; ++ks) {
      const int koff = ks * 4 + khalf * 2;  // 16x16x4 f32 layout: lane=(k/2)*16+m, vgpr=k%2
      const v2f a = *(const v2f*)&As[cur][wave * 16 + m][koff];
      v2f bfrag[8];
#pragma unroll
      for (int t = 0; t < 8; ++t) bfrag[t] = *(const v2f*)&Bs[cur][t * 16 + m][koff];
#pragma unroll
      for (int t = 0; t < 8; ++t) {
        acc[t] = __builtin_amdgcn_wmma_f32_16x16x4_f32(
            false, a, false, bfrag[t], (short)0, acc[t], false, false);
      }
    }

    if (c + 1 < DDIM / KCH) {
      asm volatile("s_wait_asynccnt 0" ::: "memory");  // my DMAs for next buf done
      __syncthreads();                                 // everyone's DMAs + reads done
    }
  }

  // Epilogue: d2 = sq_i + sq_j - 2*dot, g = exp(-d2/2), sum off-diagonal
  const float w2 = (bi == bj) ? 1.0f : 2.0f;
  float lsum = 0.f;
#pragma unroll
  for (int t = 0; t < 8; ++t) {
#pragma unroll
    for (int v = 0; v < 8; ++v) {
      const int ml = v + 8 * khalf;            // C/D layout: M = v + 8*(lane/16)
      const int i  = rowBase + wave * 16 + ml;
      const int j  = colBase + t * 16 + m;     // N = lane%16
      float d2 = sqI[wave * 16 + ml] + sqJ[t * 16 + m] - 2.0f * acc[t][v];
      d2 = fmaxf(d2, 0.0f);
      const float g = expf(-0.5f * d2);
      if (i != j) lsum += g;
    }
  }
  lsum *= w2;
#pragma unroll
  for (int mm = 16; mm >= 1; mm >>= 1) lsum += __shfl_xor(lsum, mm, 32);
  if (lane == 0) wsum[wave] = lsum;
  __syncthreads();
  if (tid == 0) {
    float s = 0.f;
    for (int w = 0; w < 8; ++w) s += wsum[w];
    neg_partial[pidx] = s;
  }
}

// ---------------------------------------------------------------------------
// Kernel 3: deterministic final reduction + scalar combine
// ---------------------------------------------------------------------------
__global__ __launch_bounds__(256) void fmicl_finalize(
    const float* __restrict__ pos_partial, int np,
    const float* __restrict__ neg_partial, int nn,
    float* __restrict__ out) {
  __shared__ float sh[256];
  const int tid = threadIdx.x;

  float p = 0.f;
  for (int i = tid; i < np; i += 256) p += pos_partial[i];
  sh[tid] = p;
  __syncthreads();
  for (int s = 128; s > 0; s >>= 1) {
    if (tid < s) sh[tid] += sh[tid + s];
    __syncthreads();
  }
  const float P = sh[0];
  __syncthreads();

  float q = 0.f;
  for (int i = tid; i < nn; i += 256) q += neg_partial[i];
  sh[tid] = q;
  __syncthreads();
  for (int s = 128; s > 0; s >>= 1) {
    if (tid < s) sh[tid] += sh[tid + s];
    __syncthreads();
  }
  if (tid == 0) {
    const float Q = sh[0];
    // star_neg = exp(log(g+eps)+1-1) = g + eps  =>  mean = Q/(N(N-1)) + eps
    const float star_mean = Q / ((float)NROWS * (float)(NROWS - 1)) + 1e-8f;
    out[0] = -(P / (float)NROWS) + 1.5f * star_mean;
  }
}

extern "C" void kernel_launch(void* const* d_in, const int* in_sizes, int n_in,
                              void* d_out, int out_size, void* d_ws, size_t ws_size,
                              hipStream_t stream) {
  const float* z1 = (const float*)d_in[0];
  const float* z2 = (const float*)d_in[1];
  float* out = (float*)d_out;

  float* ws          = (float*)d_ws;
  float* z1n         = ws;                               // N*D floats (4 MB)
  float* sq          = z1n + (size_t)NROWS * DDIM;       // N floats
  float* pos_partial = sq + NROWS;                       // 1024 floats
  float* neg_partial = pos_partial + 1024;               // 4096 floats

  fmicl_norm_pos<<<NROWS / 8, 256, 0, stream>>>(z1, z2, z1n, sq, pos_partial);
  dim3 g2(64, 64);
  fmicl_gram_neg<<<g2, 256, SMEM_BYTES, stream>>>(z1n, sq, neg_partial);
  fmicl_finalize<<<1, 256, 0, stream>>>(pos_partial, 1024, neg_partial, 4096, out);
}